// DifferentialLayer_32006096290010
// MI455X (gfx1250) — compile-verified
//
#include <hip/hip_runtime.h>

#define D        64
#define TILE     256
#define THREADS  256
#define MAXGRID  1024

typedef unsigned int u32x4 __attribute__((ext_vector_type(4)));
typedef int          i32x8 __attribute__((ext_vector_type(8)));
typedef int          i32x4 __attribute__((ext_vector_type(4)));

#if __has_builtin(__builtin_amdgcn_tensor_load_to_lds)
#define HAS_TDM 1
#else
#define HAS_TDM 0
#endif

__device__ __forceinline__ unsigned lds_lo32(const void* p) {
  // Generic (flat) pointers to LDS objects carry the wave-relative LDS byte
  // offset in their low 32 bits (ISA 10.2 aperture mapping).
  return (unsigned)(unsigned long long)p;
}

__device__ __forceinline__ void tdm_wait0() {
#if __has_builtin(__builtin_amdgcn_s_wait_tensorcnt)
  __builtin_amdgcn_s_wait_tensorcnt(0);
#else
  asm volatile("s_wait_tensorcnt 0x0" ::: "memory");
#endif
}

#if HAS_TDM
// 1D TDM copy: nwords DWORDs from global -> LDS (wave-level, TENSORcnt-tracked).
// This toolchain (clang-23 / therock) exposes the 6-arg builtin:
//   (uint32x4 g0, int32x8 g1, int32x4 g2, int32x4 g3, int32x8 extra, i32 cpol)
__device__ __forceinline__ void tdm_load_words(unsigned lds, const void* g, int nwords) {
  unsigned long long ga = (unsigned long long)g;
  u32x4 g0 = {
    1u,                                                  // count=1, is_restore=0
    lds,                                                 // lds_addr [63:32]
    (unsigned)ga,                                        // global_addr[31:0]
    ((unsigned)(ga >> 32) & 0x01FFFFFFu) | 0x80000000u   // global_addr[56:32] | type=2
  };
  unsigned nw = (unsigned)nwords;
  i32x8 g1 = {
    (int)(2u << 16),                 // workgroup_mask=0, data_size=2 (4B)
    (int)((nw & 0xFFFFu) << 16),     // tensor_dim0 lo16 @ bits[63:48]
    (int)((nw >> 16) | (1u << 16)),  // tensor_dim0 hi16; tensor_dim1=1
    (int)((nw & 0xFFFFu) << 16),     // tensor_dim1 hi=0; tile_dim0 @ bits[127:112]
    0,                               // tile_dim1=0 (unused), tile_dim2=0
    (int)nw,                         // tensor_dim0_stride lo32
    0, 0
  };
  i32x4 z4 = {0, 0, 0, 0};
  i32x8 z8 = {0, 0, 0, 0, 0, 0, 0, 0};
  __builtin_amdgcn_tensor_load_to_lds(g0, g1, z4, z4, z8, 0);
}
#endif

__device__ __forceinline__ void stage_tile(const int* __restrict__ src_idx,
                                           const int* __restrict__ dst_idx,
                                           const float* __restrict__ e_att,
                                           int tile, int E,
                                           int* ls, int* ld, float* la, int lane) {
  int base = tile * TILE;
  int cnt = E - base; if (cnt > TILE) cnt = TILE;
#if HAS_TDM
  (void)lane;
  tdm_load_words(lds_lo32(ls), src_idx + base, cnt);
  tdm_load_words(lds_lo32(ld), dst_idx + base, cnt);
  tdm_load_words(lds_lo32(la), e_att  + base, cnt);
#else
  for (int i = lane; i < cnt; i += 32) {
    ls[i] = src_idx[base + i];
    ld[i] = dst_idx[base + i];
    la[i] = e_att[base + i];
  }
#endif
}

__global__ __launch_bounds__(THREADS)
void edge_scatter_kernel(const float* __restrict__ src_emb,
                         const float* __restrict__ e_att,
                         const int*   __restrict__ src_idx,
                         const int*   __restrict__ dst_idx,
                         float*       __restrict__ out,
                         int E) {
  __shared__ int   s_src[2][TILE];
  __shared__ int   s_dst[2][TILE];
  __shared__ float s_att[2][TILE];

  const int ntiles = (E + TILE - 1) / TILE;
  const int lane = threadIdx.x & 31;
  const int wv   = threadIdx.x >> 5;

  int t   = blockIdx.x;
  int buf = 0;

  if (t < ntiles && threadIdx.x < 32) {
    stage_tile(src_idx, dst_idx, e_att, t, E, s_src[0], s_dst[0], s_att[0], lane);
    tdm_wait0();
  }
  __syncthreads();

  while (t < ntiles) {
    const int tn = t + gridDim.x;
    // Prefetch next tile's metadata into the other LDS buffer (TDM, async).
    if (tn < ntiles && threadIdx.x < 32)
      stage_tile(src_idx, dst_idx, e_att, tn, E,
                 s_src[buf ^ 1], s_dst[buf ^ 1], s_att[buf ^ 1], lane);

    // Process tile t: wave wv handles 32 consecutive edges, one edge per step.
    const int base = t * TILE;
    int cnt = E - base; if (cnt > TILE) cnt = TILE;
    const int jbeg = wv * 32;
    int jend = jbeg + 32; if (jend > cnt) jend = cnt;

    for (int j = jbeg; j < jend; ++j) {
      const int   s = __builtin_amdgcn_readfirstlane(s_src[buf][j]);
      const int   d = __builtin_amdgcn_readfirstlane(s_dst[buf][j]);
      const float a = __uint_as_float(
          __builtin_amdgcn_readfirstlane(__float_as_uint(s_att[buf][j])));

      if (j + 1 < jend) {  // hide the random-gather latency of the next row
        const int sn = __builtin_amdgcn_readfirstlane(s_src[buf][j + 1]);
        __builtin_prefetch(src_emb + (size_t)sn * D, 0, 3);
      }

      const float2 v = *(const float2*)(src_emb + (size_t)s * D + lane * 2);
      float* o = out + (size_t)d * D + lane * 2;
      (void)__hip_atomic_fetch_add(o,     v.x * a, __ATOMIC_RELAXED, __HIP_MEMORY_SCOPE_AGENT);
      (void)__hip_atomic_fetch_add(o + 1, v.y * a, __ATOMIC_RELAXED, __HIP_MEMORY_SCOPE_AGENT);
    }

    __syncthreads();                                   // everyone done with buf
    if (tn < ntiles && threadIdx.x < 32) tdm_wait0();  // next tile landed in LDS
    __syncthreads();                                   // publish buf^1
    buf ^= 1;
    t = tn;
  }
}

extern "C" void kernel_launch(void* const* d_in, const int* in_sizes, int n_in,
                              void* d_out, int out_size, void* d_ws, size_t ws_size,
                              hipStream_t stream) {
  (void)n_in; (void)d_ws; (void)ws_size;
  const float* src_emb = (const float*)d_in[0];
  const float* e_att   = (const float*)d_in[1];
  const int*   src_idx = (const int*)d_in[2];
  const int*   dst_idx = (const int*)d_in[3];
  float*       out     = (float*)d_out;

  const int E = in_sizes[2];

  (void)hipMemsetAsync(d_out, 0, (size_t)out_size * sizeof(float), stream);

  int ntiles = (E + TILE - 1) / TILE;
  int grid = ntiles < MAXGRID ? ntiles : MAXGRID;
  if (grid < 1) grid = 1;
  edge_scatter_kernel<<<grid, THREADS, 0, stream>>>(src_emb, e_att, src_idx, dst_idx, out, E);
}